// PatchTST_backbone_69071664054982
// MI455X (gfx1250) — compile-verified
//
#include <hip/hip_runtime.h>
#include <math.h>

typedef __attribute__((ext_vector_type(2))) float v2f;
typedef __attribute__((ext_vector_type(8))) float v8f;

#define TS 132          // padded LDS row stride (128 + 4) -> conflict-free WMMA operand loads
#define LN_EPS 1e-5f
#define PI_F 3.14159265358979323846f

// contiguous 8B-aligned LDS pair -> single ds_load_b64
__device__ __forceinline__ v2f lds2(const float* p) { return *(const v2f*)p; }

// ---------------- shared memory layout (~100 KB of the 320KB WGP LDS) ----------
struct SMem {
    float qt[16 * TS];      // q tile   [s][e], rows 12..15 zero
    float kt[16 * TS];      // k tile (transposed on load, then IMS'd in place)
    float vt[16 * TS];      // v tile (IMS'd in place)
    float fqre[16 * TS];    // DFT(q) real   [bin][e]   (rows 6..15 unused)
    float fqim[16 * TS];    // DFT(q) imag
    float fkre[16 * TS];    // DFT(k') real
    float fkim[16 * TS];    // DFT(k') imag
    float fkimn[16 * TS];   // -DFT(k') imag (f32 WMMA has no operand negate)
    float ybuf[12 * TS];    // IMS scratch: rows 0..8 = conv outs, rows 9..11 = stage input
    float tout[16 * TS];    // time_out = softmax(scores) @ v'
    float sc[4 * 512];      // per-wave partial 16x16 WMMA tiles (re | im halves)
    float P[256];           // softmax probabilities, padded with zeros
    float fre[6 * TS];      // freq_out real  [bin][e]
    float fim[6 * TS];      // freq_out imag
    float red[24 * 4];      // cross-wave layernorm reduction scratch
    float stats[24];        // layernorm mean / rstd
    float attw[36 * 2];     // tanh'd complex frequency attention 6x6
    float TT[12 * 6 * 2];   // folded irfft(n=6) + ft_w operator: [s][bin][re/im]
    float c12[12], s12[12]; // base 12-point twiddles (for TT)
    float ctq[72], stq[72]; // per-(bin,s) DFT twiddles for q bins
    float ctk[72], stk[72]; // per-(bin,s) DFT twiddles for k/v bins
};

__device__ __forceinline__ float wred32(float v) {
#pragma unroll
    for (int m = 16; m > 0; m >>= 1) v += __shfl_xor(v, m, 32);
    return v;
}

// ---------------- InteractiveMultiScale, fully in LDS, tile updated in place ----
__device__ void ims_inplace(float* tile, SMem& sm,
                            const float* w1, const float* b1,
                            const float* g1, const float* bb1,
                            const float* msw, const float* msb,
                            const float* we, const float* be,
                            const float* g2, const float* bb2, int tid) {
    const int e = tid, wave = tid >> 5, lane = tid & 31;

    // 1x1 conv S(12) -> INTER(3)
    float y0[3];
#pragma unroll
    for (int c = 0; c < 3; c++) {
        float acc = b1[c];
#pragma unroll
        for (int s = 0; s < 12; s++) acc += w1[c * 12 + s] * tile[s * TS + e];
        y0[c] = acc;
    }
    // LayerNorm over E per channel
#pragma unroll
    for (int c = 0; c < 3; c++) {
        float s1 = wred32(y0[c]);
        float s2 = wred32(y0[c] * y0[c]);
        if (lane == 0) { sm.red[(2 * c) * 4 + wave] = s1; sm.red[(2 * c + 1) * 4 + wave] = s2; }
    }
    __syncthreads();
    if (tid < 3) {
        float s1 = 0.f, s2 = 0.f;
        for (int w = 0; w < 4; w++) { s1 += sm.red[(2 * tid) * 4 + w]; s2 += sm.red[(2 * tid + 1) * 4 + w]; }
        float m = s1 * (1.f / 128.f), v = s2 * (1.f / 128.f) - m * m;
        sm.stats[2 * tid] = m; sm.stats[2 * tid + 1] = rsqrtf(v + LN_EPS);
    }
    __syncthreads();
#pragma unroll
    for (int c = 0; c < 3; c++) {
        float val = (y0[c] - sm.stats[2 * c]) * sm.stats[2 * c + 1] * g1[e] + bb1[e];
        sm.ybuf[(9 + c) * TS + e] = fmaxf(val, 0.f);
    }
    __syncthreads();

    // 3 sequential k=3 convs (pad=1), collecting every stage into rows 0..8
    for (int i = 0; i < 3; i++) {
        const int src = (i == 0) ? 9 : (i - 1) * 3;
        float outv[3];
#pragma unroll
        for (int o = 0; o < 3; o++) {
            float acc = msb[i * 3 + o];
#pragma unroll
            for (int ci = 0; ci < 3; ci++) {
                const float* row = &sm.ybuf[(src + ci) * TS];
                float xm = (e > 0)   ? row[e - 1] : 0.f;
                float x0 = row[e];
                float xp = (e < 127) ? row[e + 1] : 0.f;
                const float* wp = &msw[((i * 3 + o) * 3 + ci) * 3];
                acc += wp[0] * xm + wp[1] * x0 + wp[2] * xp;
            }
            outv[o] = acc;
        }
#pragma unroll
        for (int o = 0; o < 3; o++) sm.ybuf[(i * 3 + o) * TS + e] = outv[o];
        __syncthreads();
    }

    // 1x1 conv 9 -> 12, LN, relu, residual
    float y2[12];
#pragma unroll
    for (int s = 0; s < 12; s++) {
        float acc = be[s];
#pragma unroll
        for (int c = 0; c < 9; c++) acc += we[s * 9 + c] * sm.ybuf[c * TS + e];
        y2[s] = acc;
    }
#pragma unroll
    for (int s = 0; s < 12; s++) {
        float s1 = wred32(y2[s]);
        float s2 = wred32(y2[s] * y2[s]);
        if (lane == 0) { sm.red[(2 * s) * 4 + wave] = s1; sm.red[(2 * s + 1) * 4 + wave] = s2; }
    }
    __syncthreads();
    if (tid < 12) {
        float s1 = 0.f, s2 = 0.f;
        for (int w = 0; w < 4; w++) { s1 += sm.red[(2 * tid) * 4 + w]; s2 += sm.red[(2 * tid + 1) * 4 + w]; }
        float m = s1 * (1.f / 128.f), v = s2 * (1.f / 128.f) - m * m;
        sm.stats[2 * tid] = m; sm.stats[2 * tid + 1] = rsqrtf(v + LN_EPS);
    }
    __syncthreads();
#pragma unroll
    for (int s = 0; s < 12; s++) {
        float val = (y2[s] - sm.stats[2 * s]) * sm.stats[2 * s + 1] * g2[e] + bb2[e];
        tile[s * TS + e] += fmaxf(val, 0.f);
    }
    __syncthreads();
}

// ---------------- one block per (b,h); 128 threads = 4 wave32 -----------------
__global__ __launch_bounds__(128)
void fedattn_fused(const float* __restrict__ q, const float* __restrict__ k,
                   const float* __restrict__ v, const float* __restrict__ scale_p,
                   const float* __restrict__ w1, const float* __restrict__ b1,
                   const float* __restrict__ g1, const float* __restrict__ bb1,
                   const float* __restrict__ msw, const float* __restrict__ msb,
                   const float* __restrict__ we, const float* __restrict__ be,
                   const float* __restrict__ g2, const float* __restrict__ bb2,
                   const float* __restrict__ ftw, const float* __restrict__ ftb,
                   const float* __restrict__ lm1, const float* __restrict__ lm2,
                   const int* __restrict__ idxq, const int* __restrict__ idxkv,
                   float* __restrict__ out, int H) {
    __shared__ SMem sm;
    const int tid = threadIdx.x, lane = tid & 31, wave = tid >> 5;
    const int bh = blockIdx.x;
    const int h = bh % H;
    const size_t base = (size_t)bh * (12 * 128);

    // twiddle tables (computed once; LDS-broadcast afterwards)
    if (tid < 12) {
        float th = (2.f * PI_F / 12.f) * (float)tid;
        sm.c12[tid] = cosf(th);
        sm.s12[tid] = sinf(th);
    }
    if (tid < 72) {
        int j = tid / 12, s = tid % 12;
        int mq = ((idxq[j] % 12) + 12) % 12;
        int mk = ((idxkv[j] % 12) + 12) % 12;
        float thq = (2.f * PI_F / 12.f) * (float)((mq * s) % 12);
        float thk = (2.f * PI_F / 12.f) * (float)((mk * s) % 12);
        sm.ctq[tid] = cosf(thq); sm.stq[tid] = sinf(thq);
        sm.ctk[tid] = cosf(thk); sm.stk[tid] = sinf(thk);
    }
    // coalesced tile loads; k is [E,S] -> transpose-scatter into [s][e]
    const float* qb = q + base;
    const float* vb = v + base;
    const float* kb = k + base;
    for (int i = tid; i < 1536; i += 128) {
        int s = i >> 7, e = i & 127;
        sm.qt[s * TS + e] = qb[i];
        sm.vt[s * TS + e] = vb[i];
        int ke = i / 12, ks = i % 12;
        sm.kt[ks * TS + ke] = kb[i];
    }
#pragma unroll
    for (int r = 12; r < 16; r++) {
        sm.qt[r * TS + tid] = 0.f; sm.kt[r * TS + tid] = 0.f; sm.vt[r * TS + tid] = 0.f;
    }
    __syncthreads();

    // IMS on k and v (in place)
    ims_inplace(sm.kt, sm, w1, b1, g1, bb1, msw, msb, we, be, g2, bb2, tid);
    ims_inplace(sm.vt, sm, w1, b1, g1, bb1, msw, msb, we, be, g2, bb2, tid);

    // 12-point DFT over S for the selected bins; each thread owns column e.
    // q/k spectra go straight to LDS tiles (bin-major) for the WMMA contraction;
    // only the v spectrum stays in registers.
    float fvr[6] = {0.f, 0.f, 0.f, 0.f, 0.f, 0.f};
    float fvi[6] = {0.f, 0.f, 0.f, 0.f, 0.f, 0.f};
    {
        const int e = tid;
        float fqr[6] = {0.f, 0.f, 0.f, 0.f, 0.f, 0.f};
        float fqi[6] = {0.f, 0.f, 0.f, 0.f, 0.f, 0.f};
        float fkr[6] = {0.f, 0.f, 0.f, 0.f, 0.f, 0.f};
        float fki[6] = {0.f, 0.f, 0.f, 0.f, 0.f, 0.f};
#pragma unroll
        for (int s = 0; s < 12; s++) {
            float qv = sm.qt[s * TS + e];
            float kv = sm.kt[s * TS + e];
            float vv = sm.vt[s * TS + e];
#pragma unroll
            for (int j = 0; j < 6; j++) {
                float cq = sm.ctq[j * 12 + s], sq = sm.stq[j * 12 + s];
                float ck = sm.ctk[j * 12 + s], sk = sm.stk[j * 12 + s];
                fqr[j] += qv * cq; fqi[j] -= qv * sq;
                fkr[j] += kv * ck; fki[j] -= kv * sk;
                fvr[j] += vv * ck; fvi[j] -= vv * sk;
            }
        }
#pragma unroll
        for (int j = 0; j < 6; j++) {
            sm.fqre[j * TS + e] = fqr[j];
            sm.fqim[j * TS + e] = fqi[j];
            sm.fkre[j * TS + e] = fkr[j];
            sm.fkim[j * TS + e] = fki[j];
            sm.fkimn[j * TS + e] = -fki[j];
        }
    }
    __syncthreads();

    // attn_freq[s,d] = sum_e fq[s,e]*fk[d,e] (complex, no conj) as WMMA GEMMs:
    //   Re = fqr@fkr^T + fqi@(-fki)^T ;  Im = fqr@fki^T + fqi@fkr^T
    {
        const int M = lane & 15, N = lane & 15;
        const int hi = (lane >= 16);
        const int o1 = hi ? 2 : 0;
        v8f cre = {0.f, 0.f, 0.f, 0.f, 0.f, 0.f, 0.f, 0.f};
        v8f cim = {0.f, 0.f, 0.f, 0.f, 0.f, 0.f, 0.f, 0.f};
        const int kb = wave * 32;
#pragma unroll
        for (int u = 0; u < 8; u++) {
            int k0 = kb + u * 4;
            v2f ar  = lds2(&sm.fqre[M * TS + k0 + o1]);   // single ds_load_b64 each
            v2f ai  = lds2(&sm.fqim[M * TS + k0 + o1]);
            v2f br  = lds2(&sm.fkre[N * TS + k0 + o1]);
            v2f bi  = lds2(&sm.fkim[N * TS + k0 + o1]);
            v2f bin = lds2(&sm.fkimn[N * TS + k0 + o1]);
            cre = __builtin_amdgcn_wmma_f32_16x16x4_f32(false, ar, false, br,  (short)0, cre, false, false);
            cre = __builtin_amdgcn_wmma_f32_16x16x4_f32(false, ai, false, bin, (short)0, cre, false, false);
            cim = __builtin_amdgcn_wmma_f32_16x16x4_f32(false, ar, false, bi,  (short)0, cim, false, false);
            cim = __builtin_amdgcn_wmma_f32_16x16x4_f32(false, ai, false, br,  (short)0, cim, false, false);
        }
#pragma unroll
        for (int r = 0; r < 8; r++) {
            sm.sc[wave * 512 +       (r + (hi ? 8 : 0)) * 16 + (lane & 15)] = cre[r];
            sm.sc[wave * 512 + 256 + (r + (hi ? 8 : 0)) * 16 + (lane & 15)] = cim[r];
        }
    }
    __syncthreads();
    if (tid < 36) {
        float sr = 0.f, si = 0.f;
        for (int w = 0; w < 4; w++) {
            sr += sm.sc[w * 512 + (tid / 6) * 16 + (tid % 6)];
            si += sm.sc[w * 512 + 256 + (tid / 6) * 16 + (tid % 6)];
        }
        sm.attw[tid * 2] = tanhf(sr);
        sm.attw[tid * 2 + 1] = tanhf(si);
    }
    __syncthreads();

    // freq_out[j,e] = sum_d attw[j,d] * fv[d,e] (complex, K=6: scalar is cheapest)
    {
        const int e = tid;
#pragma unroll
        for (int j = 0; j < 6; j++) {
            float orr = 0.f, oi = 0.f;
#pragma unroll
            for (int d = 0; d < 6; d++) {
                float wr = sm.attw[(j * 6 + d) * 2], wi = sm.attw[(j * 6 + d) * 2 + 1];
                orr += wr * fvr[d] - wi * fvi[d];
                oi  += wr * fvi[d] + wi * fvr[d];
            }
            sm.fre[j * TS + e] = orr;
            sm.fim[j * TS + e] = oi;
        }
    }
    // fold scatter(index_q) + irfft(n=6) + (@ ft_w^T) into a 12x6 re/im operator TT
    if (tid < 12) {
        const int s = tid;
        for (int j = 0; j < 6; j++) {
            int pos = idxq[j];
            float tr = 0.f, ti = 0.f;
            if (pos >= 0 && pos <= 3) {        // irfft(n=6) uses only bins 0..3
                for (int t = 0; t < 6; t++) {
                    float w = ftw[s * 6 + t];
                    int r = (2 * pos * t) % 12;                 // angle 2*pi*pos*t/6
                    float cf = sm.c12[r], sf = sm.s12[r];
                    if (pos == 0)       tr += w * (1.f / 6.f);
                    else if (pos == 3)  tr += w * (1.f / 6.f) * cf;
                    else              { tr += w * (2.f / 6.f) * cf; ti -= w * (2.f / 6.f) * sf; }
                }
            }
            sm.TT[(s * 6 + j) * 2] = tr;
            sm.TT[(s * 6 + j) * 2 + 1] = ti;
        }
    }
    // scores = q @ k'^T via WMMA; K=128 split over the 4 waves (sc reused)
    {
        const int M = lane & 15, N = lane & 15;
        const int hi = (lane >= 16);
        const int o1 = hi ? 2 : 0;
        v8f c8 = {0.f, 0.f, 0.f, 0.f, 0.f, 0.f, 0.f, 0.f};
        const int kb = wave * 32;
#pragma unroll
        for (int u = 0; u < 8; u++) {
            int k0 = kb + u * 4;
            v2f a = lds2(&sm.qt[M * TS + k0 + o1]);
            v2f b = lds2(&sm.kt[N * TS + k0 + o1]);       // B = k'^T (contiguous along K)
            c8 = __builtin_amdgcn_wmma_f32_16x16x4_f32(false, a, false, b, (short)0, c8, false, false);
        }
#pragma unroll
        for (int r = 0; r < 8; r++)
            sm.sc[wave * 256 + (r + (hi ? 8 : 0)) * 16 + (lane & 15)] = c8[r];
    }
    __syncthreads();

    // softmax over the 12 valid columns; zero-pad P so padded WMMA lanes are inert
    if (tid < 16) {
        int s = tid;
        if (s < 12) {
            float scv = scale_p[0];
            float row[12]; float mx = -1e30f;
#pragma unroll
            for (int n = 0; n < 12; n++) {
                float x = (sm.sc[0 * 256 + s * 16 + n] + sm.sc[1 * 256 + s * 16 + n] +
                           sm.sc[2 * 256 + s * 16 + n] + sm.sc[3 * 256 + s * 16 + n]) * scv;
                row[n] = x; mx = fmaxf(mx, x);
            }
            float sum = 0.f;
#pragma unroll
            for (int n = 0; n < 12; n++) { row[n] = __expf(row[n] - mx); sum += row[n]; }
            float inv = 1.f / sum;
#pragma unroll
            for (int n = 0; n < 16; n++) sm.P[s * 16 + n] = (n < 12) ? row[n] * inv : 0.f;
        } else {
#pragma unroll
            for (int n = 0; n < 16; n++) sm.P[s * 16 + n] = 0.f;
        }
    }
    __syncthreads();

    // time_out = P @ v'  (16x128, K=16): each wave owns two 16-wide N tiles
    {
        const int M = lane & 15, N = lane & 15;
        const int hi = (lane >= 16);
        const int o1 = hi ? 2 : 0, o2 = hi ? 3 : 1;
#pragma unroll
        for (int t = 0; t < 2; t++) {
            const int n0 = (wave * 2 + t) * 16;
            v8f acc = {0.f, 0.f, 0.f, 0.f, 0.f, 0.f, 0.f, 0.f};
#pragma unroll
            for (int u = 0; u < 4; u++) {
                int k0 = u * 4;
                v2f a = lds2(&sm.P[M * 16 + k0 + o1]);    // contiguous along K
                v2f b;                                    // B rows stride TS: two b32 loads
                b.x = sm.vt[(k0 + o1) * TS + n0 + N];
                b.y = sm.vt[(k0 + o2) * TS + n0 + N];
                acc = __builtin_amdgcn_wmma_f32_16x16x4_f32(false, a, false, b, (short)0, acc, false, false);
            }
#pragma unroll
            for (int r = 0; r < 8; r++)
                sm.tout[(r + (hi ? 8 : 0)) * TS + n0 + (lane & 15)] = acc[r];
        }
    }
    __syncthreads();

    // fow[o,j] = sum_e freq_out[j,e] * (lm1 + i*lm2)[h,e,o,j]; lm stays L2-resident
    float fwr[6] = {0.f, 0.f, 0.f, 0.f, 0.f, 0.f};
    float fwi[6] = {0.f, 0.f, 0.f, 0.f, 0.f, 0.f};
    {
        const int o = tid;
        const float* p1 = lm1 + ((size_t)h * 128 * 128 + (size_t)o) * 6;
        const float* p2 = lm2 + ((size_t)h * 128 * 128 + (size_t)o) * 6;
        for (int e = 0; e < 128; e++) {
            const float2* l1 = (const float2*)(p1 + (size_t)e * (128 * 6));
            const float2* l2 = (const float2*)(p2 + (size_t)e * (128 * 6));
            float2 a01 = l1[0], a23 = l1[1], a45 = l1[2];
            float2 b01 = l2[0], b23 = l2[1], b45 = l2[2];
            float la[6] = {a01.x, a01.y, a23.x, a23.y, a45.x, a45.y};
            float lb[6] = {b01.x, b01.y, b23.x, b23.y, b45.x, b45.y};
#pragma unroll
            for (int j = 0; j < 6; j++) {
                float fr = sm.fre[j * TS + e], fi = sm.fim[j * TS + e];   // LDS broadcast
                fwr[j] += fr * la[j] - fi * lb[j];
                fwi[j] += fr * lb[j] + fi * la[j];
            }
        }
    }

    // out[b,h,s,e] = time_out + scale_freq * (ft_b[s] + TT[s,:] . fow[e,:])
    {
        const int e = tid;
        const float sfq = 1.f / (128.f * 128.f);
        float* ob = out + base;
#pragma unroll
        for (int s = 0; s < 12; s++) {
            float acc = ftb[s];
#pragma unroll
            for (int j = 0; j < 6; j++)
                acc += sm.TT[(s * 6 + j) * 2] * fwr[j] + sm.TT[(s * 6 + j) * 2 + 1] * fwi[j];
            ob[s * 128 + e] = sm.tout[s * TS + e] + acc * sfq;
        }
    }
}

extern "C" void kernel_launch(void* const* d_in, const int* in_sizes, int n_in,
                              void* d_out, int out_size, void* d_ws, size_t ws_size,
                              hipStream_t stream) {
    const float* q    = (const float*)d_in[0];
    const float* k    = (const float*)d_in[1];
    const float* v    = (const float*)d_in[2];
    const float* scal = (const float*)d_in[3];
    const float* w1   = (const float*)d_in[4];
    const float* b1   = (const float*)d_in[5];
    const float* g1   = (const float*)d_in[6];
    const float* bb1  = (const float*)d_in[7];
    const float* msw  = (const float*)d_in[8];
    const float* msb  = (const float*)d_in[9];
    const float* we   = (const float*)d_in[10];
    const float* be   = (const float*)d_in[11];
    const float* g2   = (const float*)d_in[12];
    const float* bb2  = (const float*)d_in[13];
    const float* ftw  = (const float*)d_in[14];
    const float* ftb  = (const float*)d_in[15];
    const float* lm1  = (const float*)d_in[16];
    const float* lm2  = (const float*)d_in[17];
    const int*   idxq = (const int*)d_in[18];
    const int*   idxkv= (const int*)d_in[19];

    const int BH = in_sizes[0] / (12 * 128);          // B*H = 16384
    const int H  = in_sizes[16] / (128 * 128 * 6);    // from lm1 [H,E,E,NSEL]

    fedattn_fused<<<BH, 128, 0, stream>>>(q, k, v, scal, w1, b1, g1, bb1, msw, msb,
                                          we, be, g2, bb2, ftw, ftb, lm1, lm2,
                                          idxq, idxkv, (float*)d_out, H);
    (void)d_ws; (void)ws_size; (void)n_in; (void)out_size;
}